// QuantizationLayer_36721970380837
// MI455X (gfx1250) — compile-verified
//
#include <hip/hip_runtime.h>
#include <stdint.h>

#define Hh 180
#define Ww 240
#define Bb 64
#define Nn 131072
#define Ss 32
#define SEGL 4096      // Nn / Ss
#define STARTI 2
#define VWORDS 352     // ceil(120*90 / 32) = 338, padded

typedef __attribute__((ext_vector_type(2))) float v2f;
typedef __attribute__((ext_vector_type(8))) float v8f;

// ---------------------------------------------------------------------------
// Kernel A: per-(batch,segment) histograms in LDS, pack coords, f64 sums for
// the final recenter means. One block per (b,s), 256 threads, 16 events each.
// ---------------------------------------------------------------------------
__global__ __launch_bounds__(256)
void k_hist(const float* __restrict__ ev, uint32_t* __restrict__ coords,
            int* __restrict__ hx, int* __restrict__ hy,
            double* __restrict__ sumx, double* __restrict__ sumy) {
  __shared__ int shx[Ww];
  __shared__ int shy[Hh];
  __shared__ float ssx, ssy;
  const int blk = blockIdx.x;           // b*S + s
  const int b = blk / Ss;
  const int tid = threadIdx.x;
  for (int i = tid; i < Ww; i += 256) shx[i] = 0;
  for (int i = tid; i < Hh; i += 256) shy[i] = 0;
  if (tid == 0) { ssx = 0.f; ssy = 0.f; }
  __syncthreads();

  const float* base = ev + (size_t)blk * SEGL * 5;
  uint32_t* cbase = coords + (size_t)blk * SEGL;
  float lx = 0.f, ly = 0.f;
  for (int n = tid; n < SEGL; n += 256) {
    float xf = base[(size_t)n * 5 + 0];
    float yf = base[(size_t)n * 5 + 1];
    int xi = (int)xf;                   // astype(int32): truncation
    int yi = (int)yf;
    atomicAdd(&shx[xi], 1);
    atomicAdd(&shy[yi], 1);
    cbase[n] = (uint32_t)(xi & 0xFFFF) | ((uint32_t)(yi & 0xFFFF) << 16);
    lx += xf; ly += yf;
  }
  atomicAdd(&ssx, lx);
  atomicAdd(&ssy, ly);
  __syncthreads();

  int* hxo = hx + (size_t)blk * Ww;
  int* hyo = hy + (size_t)blk * Hh;
  for (int i = tid; i < Ww; i += 256) hxo[i] = shx[i];
  for (int i = tid; i < Hh; i += 256) hyo[i] = shy[i];
  if (tid == 0) {
    atomicAdd(&sumx[b], (double)ssx);  // block partials are exact integers
    atomicAdd(&sumy[b], (double)ssy);
  }
}

// ---------------------------------------------------------------------------
// Kernel B: per batch: top-2 noise peaks, zero columns (via zeroed weights),
// weighted mean per segment via V_WMMA_F32_16X16X4_F32 (exact: all integers
// < 2^24), then ax/ay = round(mean_s - mean_START).
// ---------------------------------------------------------------------------
__device__ __forceinline__ unsigned long long mkkey(int v, int idx) {
  // max value wins; on ties, smaller index wins (matches jax.lax.top_k)
  return ((unsigned long long)(unsigned)v << 32) |
         (unsigned long long)(0xFFFFFFFFu - (unsigned)idx);
}

// Weighted row-sums of a 32 x D f32 matrix against weight vector w, using
// two 16x16x4 WMMA accumulators (rows 0-15, rows 16-31). The weight is
// broadcast across all 16 N-columns of B, so the result does not depend on
// the B-matrix N layout; K layout matches the A layout (lanes 0-15: K=0,1;
// lanes 16-31: K=2,3). mean[s] = (sum_d hist[s][d]*w[d]) / SEGL.
__device__ __forceinline__ void wmma_meansum(const float* hist, int D,
                                             const float* w, float* mean,
                                             int lane) {
  const int mrow = lane & 15;
  const int khalf = (lane >> 4) << 1;   // 0 or 2
  v8f c0 = {}; v8f c1 = {};
  for (int kb = 0; kb < D; kb += 4) {
    const int k0 = kb + khalf;
    v2f a0, a1, bv;
    a0.x = hist[mrow * D + k0];        a0.y = hist[mrow * D + k0 + 1];
    a1.x = hist[(mrow + 16) * D + k0]; a1.y = hist[(mrow + 16) * D + k0 + 1];
    bv.x = w[k0];                      bv.y = w[k0 + 1];
    c0 = __builtin_amdgcn_wmma_f32_16x16x4_f32(false, a0, false, bv,
                                               (short)0, c0, false, false);
    c1 = __builtin_amdgcn_wmma_f32_16x16x4_f32(false, a1, false, bv,
                                               (short)0, c1, false, false);
  }
  // D layout: VGPR i holds M = i + 8*(lane/16), N = lane%16. All N columns
  // carry the same value; take N==0 lanes.
  if ((lane & 15) == 0) {
    const int half = lane >> 4;
    const float inv = 1.0f / (float)SEGL;
    #pragma unroll
    for (int i = 0; i < 8; i++) {
      mean[i + 8 * half]      = c0[i] * inv;
      mean[16 + i + 8 * half] = c1[i] * inv;
    }
  }
}

__global__ __launch_bounds__(256)
void k_align(const int* __restrict__ hx, const int* __restrict__ hy,
             int* __restrict__ ax, int* __restrict__ ay) {
  __shared__ float fhx[Ss * Ww];        // 30 KB
  __shared__ float fhy[Ss * Hh];        // 22.5 KB
  __shared__ float wX[Ww], wY[Hh];
  __shared__ float meanX[Ss], meanY[Ss];
  __shared__ unsigned long long key;
  __shared__ int nc[4];                 // ncx0, ncx1, ncy0, ncy1
  const int b = blockIdx.x, tid = threadIdx.x;
  const int* gx = hx + (size_t)b * Ss * Ww;
  const int* gy = hy + (size_t)b * Ss * Hh;
  for (int i = tid; i < Ss * Ww; i += 256) fhx[i] = (float)gx[i];
  for (int i = tid; i < Ss * Hh; i += 256) fhy[i] = (float)gy[i];
  __syncthreads();

  // top-2 over flattened [S*D], per axis (values are exact small ints in f32)
  int excl = -1;
  #pragma unroll
  for (int pass = 0; pass < 4; pass++) {
    const bool isX = pass < 2;
    const int total = isX ? Ss * Ww : Ss * Hh;
    const float* f = isX ? fhx : fhy;
    if (tid == 0) key = 0ull;
    if ((pass & 1) == 0) excl = -1;     // first pass per axis: no exclusion
    __syncthreads();
    unsigned long long lk = 0ull;
    for (int i = tid; i < total; i += 256)
      if (i != excl) {
        unsigned long long k2 = mkkey((int)f[i], i);
        if (k2 > lk) lk = k2;
      }
    atomicMax(&key, lk);
    __syncthreads();
    const int idx = (int)(0xFFFFFFFFu - (unsigned)(key & 0xFFFFFFFFull));
    excl = idx;
    if (tid == 0) nc[pass] = idx % (isX ? Ww : Hh);
    __syncthreads();
  }

  for (int d = tid; d < Ww; d += 256)
    wX[d] = (d == nc[0] || d == nc[1]) ? 0.f : (float)d;
  for (int d = tid; d < Hh; d += 256)
    wY[d] = (d == nc[2] || d == nc[3]) ? 0.f : (float)d;
  __syncthreads();

  const int wid = tid >> 5, lane = tid & 31;
  if (wid == 0)      wmma_meansum(fhx, Ww, wX, meanX, lane);  // EXEC all-1
  else if (wid == 1) wmma_meansum(fhy, Hh, wY, meanY, lane);  // EXEC all-1
  __syncthreads();

  if (tid < Ss) {
    ax[b * Ss + tid] = (int)rintf(meanX[tid] - meanX[STARTI]); // RNE = jnp.round
    ay[b * Ss + tid] = (int)rintf(meanY[tid] - meanY[STARTI]);
  }
}

// ---------------------------------------------------------------------------
// Kernel C: per batch: container + verifier bitmap in (dynamic) LDS,
// sequential segment scan with uniform early exit, fused recenter writeout.
// ---------------------------------------------------------------------------
__global__ __launch_bounds__(1024)
void k_scan(const uint32_t* __restrict__ coords, const int* __restrict__ ax,
            const int* __restrict__ ay, const double* __restrict__ sumx,
            const double* __restrict__ sumy, float* __restrict__ out) {
  extern __shared__ int smem[];
  int* cont = smem;                                  // H*W ints (172.8 KB)
  unsigned* ver = (unsigned*)(smem + Hh * Ww);       // VWORDS words
  int* cntp = (int*)(ver + VWORDS);
  const int b = blockIdx.x, tid = threadIdx.x;
  const int NT = 1024;

  for (int i = tid; i < Hh * Ww; i += NT) cont[i] = 0;
  for (int i = tid; i < VWORDS; i += NT) ver[i] = 0u;
  if (tid == 0) *cntp = 0;
  __syncthreads();

  const uint32_t* cb = coords + (size_t)b * Nn;
  const int* axb = ax + b * Ss;
  const int* ayb = ay + b * Ss;

  // Segment START: unconditional container add + verifier init (count bits)
  {
    const int axs = axb[STARTI], ays = ayb[STARTI];
    int local = 0;
    for (int n = tid; n < SEGL; n += NT) {
      const uint32_t c = cb[STARTI * SEGL + n];
      int xs = (int)(c & 0xFFFFu) - axs;
      int ys = (int)(c >> 16) - ays;
      xs = min(max(xs, 0), Ww - 1);
      ys = min(max(ys, 0), Hh - 1);
      atomicAdd(&cont[xs + Ww * ys], 1);
      const int iv = (xs >> 1) + (Ww / 2) * (ys >> 1);
      const unsigned bit = 1u << (iv & 31);
      const unsigned old = atomicOr(&ver[iv >> 5], bit);
      if (!(old & bit)) local++;
    }
    atomicAdd(cntp, local);
    __syncthreads();
  }
  int total = *cntp;

  for (int s = STARTI + 1; s < Ss; s++) {
    __syncthreads();
    if (tid == 0) *cntp = 0;
    __syncthreads();
    const int axs = axb[s], ays = ayb[s];
    int icl[SEGL / NT];
    int ne = 0, local = 0;
    for (int n = tid; n < SEGL; n += NT) {
      const uint32_t c = cb[s * SEGL + n];
      int xs = (int)(c & 0xFFFFu) - axs;
      int ys = (int)(c >> 16) - ays;
      xs = min(max(xs, 0), Ww - 1);
      ys = min(max(ys, 0), Hh - 1);
      icl[ne++] = xs + Ww * ys;
      const int iv = (xs >> 1) + (Ww / 2) * (ys >> 1);
      const unsigned bit = 1u << (iv & 31);
      const unsigned old = atomicOr(&ver[iv >> 5], bit);
      if (!(old & bit)) local++;
    }
    atomicAdd(cntp, local);
    __syncthreads();
    const int newc = *cntp;
    const float ratio = (float)newc / (float)(total + newc);
    if (ratio < 0.1f) break;            // uniform across block
    total += newc;
    for (int e = 0; e < ne; e++) atomicAdd(&cont[icl[e]], 1);
  }
  __syncthreads();

  // recenter shift: dx = floor(W/2 - mean_x), gather with zero fill
  const float mx = (float)(sumx[b] / (double)Nn);
  const float my = (float)(sumy[b] / (double)Nn);
  const int dx = (int)floorf((float)(Ww / 2) - mx);
  const int dy = (int)floorf((float)(Hh / 2) - my);
  float* ob = out + (size_t)b * Hh * Ww;
  for (int p = tid; p < Hh * Ww; p += NT) {
    const int i = p / Ww, j = p % Ww;
    const int ii = i - dy, jj = j - dx;
    float v = 0.f;
    if (ii >= 0 && ii < Hh && jj >= 0 && jj < Ww) v = (float)cont[ii * Ww + jj];
    ob[p] = v;
  }
}

// ---------------------------------------------------------------------------
extern "C" void kernel_launch(void* const* d_in, const int* in_sizes, int n_in,
                              void* d_out, int out_size, void* d_ws, size_t ws_size,
                              hipStream_t stream) {
  const float* ev = (const float*)d_in[0];
  char* ws = (char*)d_ws;
  size_t o = 0;
  uint32_t* coords = (uint32_t*)(ws + o); o += (size_t)Bb * Nn * 4;       // 33.6 MB
  int* hx = (int*)(ws + o);               o += (size_t)Bb * Ss * Ww * 4;  // 2.0 MB
  int* hy = (int*)(ws + o);               o += (size_t)Bb * Ss * Hh * 4;  // 1.5 MB
  int* ax = (int*)(ws + o);               o += (size_t)Bb * Ss * 4;
  int* ay = (int*)(ws + o);               o += (size_t)Bb * Ss * 4;
  double* sumx = (double*)(ws + o);       o += (size_t)Bb * 8;
  double* sumy = (double*)(ws + o);       o += (size_t)Bb * 8;

  hipMemsetAsync(sumx, 0, (size_t)Bb * 16, stream);  // zero sumx+sumy

  k_hist<<<Bb * Ss, 256, 0, stream>>>(ev, coords, hx, hy, sumx, sumy);
  k_align<<<Bb, 256, 0, stream>>>(hx, hy, ax, ay);

  const size_t shmem = (size_t)Hh * Ww * 4 + (size_t)VWORDS * 4 + 16;
  k_scan<<<Bb, 1024, shmem, stream>>>(coords, ax, ay, sumx, sumy, (float*)d_out);
}